// RecurrentResonance_59339268162185
// MI455X (gfx1250) — compile-verified
//
#include <hip/hip_runtime.h>
#include <math.h>

typedef float v2f __attribute__((ext_vector_type(2)));
typedef float v8f __attribute__((ext_vector_type(8)));

__device__ __forceinline__ v8f wmma4(v2f a, v2f b, v8f c) {
  return __builtin_amdgcn_wmma_f32_16x16x4_f32(false, a, false, b, (short)0, c, false, false);
}
__device__ __forceinline__ int waveid() {
  return __builtin_amdgcn_readfirstlane((int)(threadIdx.x >> 5));
}

// B=1024 signals, 128 frames, WIN=512, HOP=256, 257 coeffs -> 514 re|im, padded to 528.
// All GEMM B-operands live in "fragment-packed" layout: elem (k,n) at [((k>>1)*N + n)*2 + (k&1)],
// so a lane's B fragment {B[k,n], B[k+1,n]} is one aligned b64 load.
#define NCOL 528
#define PI_D 3.14159265358979323846

// ---------------- utility ----------------
__global__ void k_zero(float* __restrict__ p, int n) {
  int i = blockIdx.x * blockDim.x + threadIdx.x;
  if (i < n) p[i] = 0.f;
}

// Build packed windowed-DFT matrix Wstft (K=512 x N=528) and packed inverse matrix Dmat (K=528 x N=512)
__global__ void k_init_mats(float* __restrict__ Wstft, float* __restrict__ Dmat) {
  int i = blockIdx.x * blockDim.x + threadIdx.x;
  const int NW = 512 * NCOL;
  if (i < NW) {
    int kl = i & 1, pairIdx = i >> 1;
    int n = pairIdx % NCOL, jp = pairIdx / NCOL;
    int j = jp * 2 + kl;                                   // time sample (K dim)
    double win = 0.5 - 0.5 * cos(2.0 * PI_D * (double)j / 511.0); // numpy hanning(512)
    double v = 0.0;
    if (n < 257)      v =  win * cos(2.0 * PI_D * (double)(j * n) / 512.0);
    else if (n < 514) v = -win * sin(2.0 * PI_D * (double)(j * (n - 257)) / 512.0);
    Wstft[i] = (float)v;
  } else if (i < NW + NCOL * 512) {
    int t = i - NW;
    int kl = t & 1, pairIdx = t >> 1;
    int j = pairIdx % 512, rp = pairIdx / 512;
    int r = rp * 2 + kl;                                   // spectral row (K dim)
    double v = 0.0;
    if (r < 257) {
      double coef = (r == 0 || r == 256) ? (1.0 / 512.0) : (2.0 / 512.0);
      v = coef * cos(2.0 * PI_D * (double)(j * r) / 512.0);
    } else if (r < 514) {
      int k = r - 257;
      if (k != 0 && k != 256) v = -(2.0 / 512.0) * sin(2.0 * PI_D * (double)(j * k) / 512.0);
    }
    Dmat[t] = (float)v;
  }
}

// ---------------- weight repack kernels (into fragment-packed layout) ----------------
__global__ void k_pack_wlat(const float* __restrict__ src, float* __restrict__ dst) {
  int i = blockIdx.x * blockDim.x + threadIdx.x;       // over 256*2048
  if (i >= 256 * 2048) return;
  int k = i >> 11, n = i & 2047;
  dst[((k >> 1) * 2048 + n) * 2 + (k & 1)] = src[i];
}
__global__ void k_pack_ups(const float* __restrict__ src, float* __restrict__ dst) {
  int i = blockIdx.x * blockDim.x + threadIdx.x;       // over 4*256*256*4
  if (i >= 4 * 256 * 256 * 4) return;
  int tap = i & 3, rest = i >> 2;
  int n = rest & 255, rest2 = rest >> 8;
  int k = rest2 & 255, layer = rest2 >> 8;
  dst[(layer * 4 + tap) * 65536 + ((k >> 1) * 256 + n) * 2 + (k & 1)] = src[i];
}
__global__ void k_pack_fin(const float* __restrict__ src, float* __restrict__ dst) {
  const int PER = 128 * NCOL * 2;                      // 135168 per tap
  int i = blockIdx.x * blockDim.x + threadIdx.x;       // over 3*PER
  if (i >= 3 * PER) return;
  int tap = i / PER, r = i % PER;
  int kl = r & 1, p = r >> 1;
  int n = p % NCOL, kp = p / NCOL;
  int k = kp * 2 + kl;
  dst[i] = (n < 514) ? src[(n * 256 + k) * 3 + tap] : 0.f;  // final_w[o=n][i=k][h=tap]
}

// ---------------- STFT: [B*64,512] @ Wstft[512,528] -> spec [B*64,528] ----------------
template <bool FULL, bool MASKED>
__device__ __forceinline__ void stft_kchunk(int kbeg, int kend, int kk,
    const float* __restrict__ src, int base, const v2f* __restrict__ W2, int nb,
    v8f& acc0, v8f& acc1, v8f& acc2, v8f& acc3) {
#pragma unroll 4
  for (int k0 = kbeg; k0 < kend; k0 += 4) {
    int s0 = base + k0;                                  // even -> one mask covers both elems
    v2f a;
    if (MASKED) {
      v2f av = *(const v2f*)(src + (s0 < 16382 ? s0 : 16382));
      a = av * ((s0 < 16384) ? 1.f : 0.f);
    } else {
      a = *(const v2f*)(src + s0);
    }
    const v2f* wr = W2 + (size_t)((k0 + kk) >> 1) * NCOL + nb;
    acc0 = wmma4(a, wr[0], acc0);
    if (FULL) {
      acc1 = wmma4(a, wr[16], acc1);
      acc2 = wmma4(a, wr[32], acc2);
      acc3 = wmma4(a, wr[48], acc3);
    }
  }
}

__global__ void k_stft(const float* __restrict__ imp, const float* __restrict__ Wm,
                       float* __restrict__ spec) {
  const int NSTRIP = 9;
  int wid = blockIdx.x * 4 + waveid();
  int mTile = wid / NSTRIP, strip = wid % NSTRIP;
  if (mTile >= 4096) return;
  bool full = (strip != 8);
  bool boundary = ((mTile & 3) == 3);   // tile contains rows f in [48,63]; only f=63, k>=256 can OOB
  int l = threadIdx.x & 31;
  int kk = (l >> 4) * 2;
  int mrow = mTile * 16 + (l & 15);
  int b = mrow >> 6, f = mrow & 63;
  const float* src = imp + b * 16384;
  const v2f* W2 = (const v2f*)Wm;
  int nb = strip * 64 + (l & 15);
  int base = 256 * f + kk;
  v8f acc0 = {}, acc1 = {}, acc2 = {}, acc3 = {};
  if (full) {
    if (!boundary) {
      stft_kchunk<true, false>(0, 512, kk, src, base, W2, nb, acc0, acc1, acc2, acc3);
    } else {
      stft_kchunk<true, false>(0, 256, kk, src, base, W2, nb, acc0, acc1, acc2, acc3);
      stft_kchunk<true, true>(256, 512, kk, src, base, W2, nb, acc0, acc1, acc2, acc3);
    }
  } else {
    if (!boundary) {
      stft_kchunk<false, false>(0, 512, kk, src, base, W2, nb, acc0, acc1, acc2, acc3);
    } else {
      stft_kchunk<false, false>(0, 256, kk, src, base, W2, nb, acc0, acc1, acc2, acc3);
      stft_kchunk<false, true>(256, 512, kk, src, base, W2, nb, acc0, acc1, acc2, acc3);
    }
  }
  int mr0 = mTile * 16 + 8 * (l >> 4);
  int nc = strip * 64 + (l & 15);
#pragma unroll
  for (int v = 0; v < 8; v++) {
    float* dst = spec + (mr0 + v) * NCOL;
    dst[nc] = acc0[v];
    if (full) { dst[nc + 16] = acc1[v]; dst[nc + 32] = acc2[v]; dst[nc + 48] = acc3[v]; }
  }
}

// ---------------- latent GEMM: [1024,256]@[256,2048]+bias, scatter to [B,8,256] -------------
__global__ void k_lat(const float* __restrict__ lat, const float* __restrict__ Wp,
                      const float* __restrict__ bias, float* __restrict__ X0) {
  const int NSTRIP = 32;
  int wid = blockIdx.x * 4 + waveid();
  int mTile = wid / NSTRIP, strip = wid % NSTRIP;
  if (mTile >= 64) return;
  int l = threadIdx.x & 31, kk = (l >> 4) * 2;
  int mrow = mTile * 16 + (l & 15);
  int nb = strip * 64 + (l & 15);
  const v2f* A2 = (const v2f*)(lat + mrow * 256);
  const v2f* W2 = (const v2f*)Wp;
  v8f acc0 = {}, acc1 = {}, acc2 = {}, acc3 = {};
#pragma unroll 4
  for (int k0 = 0; k0 < 256; k0 += 4) {
    int kp = (k0 + kk) >> 1;
    v2f a = A2[kp];
    const v2f* wr = W2 + (size_t)kp * 2048 + nb;
    acc0 = wmma4(a, wr[0], acc0);
    acc1 = wmma4(a, wr[16], acc1);
    acc2 = wmma4(a, wr[32], acc2);
    acc3 = wmma4(a, wr[48], acc3);
  }
  int mr0 = mTile * 16 + 8 * (l >> 4);
#pragma unroll
  for (int v = 0; v < 8; v++) {
    int row = mr0 + v;
    int n0 = strip * 64 + (l & 15);
    // col j of [B,2048] maps to (t=j&7, c=j>>3) in [B,8,256] layout
    X0[row * 2048 + (n0 & 7) * 256 + (n0 >> 3)]           = acc0[v] + bias[n0];
    X0[row * 2048 + ((n0+16) & 7) * 256 + ((n0+16) >> 3)] = acc1[v] + bias[n0+16];
    X0[row * 2048 + ((n0+32) & 7) * 256 + ((n0+32) >> 3)] = acc2[v] + bias[n0+32];
    X0[row * 2048 + ((n0+48) & 7) * 256 + ((n0+48) >> 3)] = acc3[v] + bias[n0+48];
  }
}

// ---------------- transposed conv layer (stride 2, k=4, SAME), one parity per launch --------
// even t=2u:  Y = X[u-1]@W[h=0] + X[u]@W[h=2] ; odd t=2u+1: Y = X[u]@W[h=1] + X[u+1]@W[h=3]
template <bool MASKED>
__device__ __forceinline__ void convt_kloop(int kk,
    const v2f* __restrict__ X0v, const v2f* __restrict__ X1v, float m0, float m1,
    const v2f* __restrict__ W20, const v2f* __restrict__ W21, int nb,
    v8f& acc0, v8f& acc1, v8f& acc2, v8f& acc3) {
#pragma unroll 4
  for (int k0 = 0; k0 < 256; k0 += 4) {
    int kp = (k0 + kk) >> 1;
    v2f a0 = X0v[kp];
    v2f a1 = X1v[kp];
    if (MASKED) { a0 *= m0; a1 *= m1; }
    const v2f* wr0 = W20 + (size_t)kp * 256 + nb;
    const v2f* wr1 = W21 + (size_t)kp * 256 + nb;
    acc0 = wmma4(a0, wr0[0],  acc0);  acc0 = wmma4(a1, wr1[0],  acc0);
    acc1 = wmma4(a0, wr0[16], acc1);  acc1 = wmma4(a1, wr1[16], acc1);
    acc2 = wmma4(a0, wr0[32], acc2);  acc2 = wmma4(a1, wr1[32], acc2);
    acc3 = wmma4(a0, wr0[48], acc3);  acc3 = wmma4(a1, wr1[48], acc3);
  }
}

__global__ void k_convt(const float* __restrict__ X, const float* __restrict__ Wp0,
                        const float* __restrict__ Wp1, float* __restrict__ Y,
                        int LinLog2, int parity) {
  int Lin = 1 << LinLog2;
  int mTiles = (1024 * Lin) >> 4;
  int wid = blockIdx.x * 4 + waveid();
  int mTile = wid >> 2, strip = wid & 3;               // NSTRIP = 4 (N=256)
  if (mTile >= mTiles) return;
  int l = threadIdx.x & 31, kk = (l >> 4) * 2;
  int mrow = mTile * 16 + (l & 15);
  int b = mrow >> LinLog2, u = mrow & (Lin - 1);
  int s0 = u + (parity ? 0 : -1);
  int s1 = u + (parity ? 1 : 0);
  // scalar decision: does this tile contain an out-of-range source row?
  int tilesPerB = Lin >> 4;                            // 0 when Lin==8
  int tmod = tilesPerB ? (mTile % tilesPerB) : 0;
  bool needMask = (tilesPerB == 0) ||
                  (parity == 0 ? (tmod == 0) : (tmod == tilesPerB - 1));
  const v2f* W20 = (const v2f*)Wp0;
  const v2f* W21 = (const v2f*)Wp1;
  int nb = strip * 64 + (l & 15);
  v8f acc0 = {}, acc1 = {}, acc2 = {}, acc3 = {};
  if (needMask) {
    float m0 = (s0 >= 0 && s0 < Lin) ? 1.f : 0.f;
    float m1 = (s1 >= 0 && s1 < Lin) ? 1.f : 0.f;
    int s0c = s0 < 0 ? 0 : (s0 >= Lin ? Lin - 1 : s0);
    int s1c = s1 < 0 ? 0 : (s1 >= Lin ? Lin - 1 : s1);
    const v2f* X0v = (const v2f*)(X + (b * Lin + s0c) * 256);
    const v2f* X1v = (const v2f*)(X + (b * Lin + s1c) * 256);
    convt_kloop<true>(kk, X0v, X1v, m0, m1, W20, W21, nb, acc0, acc1, acc2, acc3);
  } else {
    const v2f* X0v = (const v2f*)(X + (b * Lin + s0) * 256);
    const v2f* X1v = (const v2f*)(X + (b * Lin + s1) * 256);
    convt_kloop<false>(kk, X0v, X1v, 1.f, 1.f, W20, W21, nb, acc0, acc1, acc2, acc3);
  }
#pragma unroll
  for (int v = 0; v < 8; v++) {
    int mr = mTile * 16 + v + 8 * (l >> 4);
    int bb = mr >> LinLog2, uu = mr & (Lin - 1);
    float* dst = Y + ((bb << (LinLog2 + 1)) + 2 * uu + parity) * 256;
    dst[nb] = acc0[v]; dst[nb + 16] = acc1[v]; dst[nb + 32] = acc2[v]; dst[nb + 48] = acc3[v];
  }
}

// ---------------- batchnorm stats + apply (mean/var over batch and time) ----------------
__global__ void k_bnreduce(const float* __restrict__ Y, float* __restrict__ stats, int nRows) {
  int c = threadIdx.x;           // 256 channels
  int row0 = blockIdx.x * 128;
  int rend = row0 + 128; if (rend > nRows) rend = nRows;
  float s = 0.f, q = 0.f;
  for (int r = row0; r < rend; r++) {
    float v = Y[r * 256 + c];
    s += v; q += v * v;
  }
  atomicAdd(&stats[c], s);
  atomicAdd(&stats[256 + c], q);
}

__global__ void k_bnapply(float* __restrict__ Y, const float* __restrict__ stats,
                          const float* __restrict__ g, const float* __restrict__ be,
                          int nElem, float invCnt) {
  int i = blockIdx.x * blockDim.x + threadIdx.x;
  if (i >= nElem) return;
  int c = i & 255;
  float m = stats[c] * invCnt;
  float var = stats[256 + c] * invCnt - m * m;
  float x = (Y[i] - m) * rsqrtf(var + 1e-5f) * g[c] + be[c];
  Y[i] = x > 0.f ? x : 0.2f * x;
}

// ---------------- final conv (k=3, SAME) -> res[B*128, 528] (gain re|im layout) ------------
template <bool FULL, bool MASKED>
__device__ __forceinline__ void fconv_kloop(int kk,
    const v2f* __restrict__ XM, const v2f* __restrict__ XC, const v2f* __restrict__ XP,
    float mm, float mp,
    const v2f* __restrict__ W0, const v2f* __restrict__ W1, const v2f* __restrict__ W2c, int nb,
    v8f& acc0, v8f& acc1, v8f& acc2, v8f& acc3) {
#pragma unroll 2
  for (int k0 = 0; k0 < 256; k0 += 4) {
    int kp = (k0 + kk) >> 1;
    v2f am = XM[kp];
    v2f a0 = XC[kp];
    v2f ap = XP[kp];
    if (MASKED) { am *= mm; ap *= mp; }
    const v2f* w0 = W0 + (size_t)kp * NCOL + nb;
    const v2f* w1 = W1 + (size_t)kp * NCOL + nb;
    const v2f* w2 = W2c + (size_t)kp * NCOL + nb;
    acc0 = wmma4(am, w0[0], acc0); acc0 = wmma4(a0, w1[0], acc0); acc0 = wmma4(ap, w2[0], acc0);
    if (FULL) {
      acc1 = wmma4(am, w0[16], acc1); acc1 = wmma4(a0, w1[16], acc1); acc1 = wmma4(ap, w2[16], acc1);
      acc2 = wmma4(am, w0[32], acc2); acc2 = wmma4(a0, w1[32], acc2); acc2 = wmma4(ap, w2[32], acc2);
      acc3 = wmma4(am, w0[48], acc3); acc3 = wmma4(a0, w1[48], acc3); acc3 = wmma4(ap, w2[48], acc3);
    }
  }
}

__global__ void k_fconv(const float* __restrict__ X, const float* __restrict__ WpF,
                        const float* __restrict__ bf, float* __restrict__ R) {
  const int NSTRIP = 9;
  const int PER2 = 128 * NCOL;                          // v2f elements per packed tap
  int wid = blockIdx.x * 4 + waveid();
  int mTile = wid / NSTRIP, strip = wid % NSTRIP;
  if (mTile >= 8192) return;
  bool full = (strip != 8);
  bool boundary = ((mTile & 7) == 0) || ((mTile & 7) == 7);  // tile contains t=0 or t=127
  int l = threadIdx.x & 31, kk = (l >> 4) * 2;
  int mrow = mTile * 16 + (l & 15);
  int b = mrow >> 7, t = mrow & 127;
  const v2f* XC = (const v2f*)(X + (b * 128 + t) * 256);
  const v2f* W0 = (const v2f*)WpF;                      // tap h=0
  const v2f* W1 = W0 + PER2;                            // tap h=1
  const v2f* W2 = W1 + PER2;                            // tap h=2
  int nb = strip * 64 + (l & 15);
  v8f acc0 = {}, acc1 = {}, acc2 = {}, acc3 = {};
  if (!boundary) {
    const v2f* XM = (const v2f*)(X + (b * 128 + t - 1) * 256);
    const v2f* XP = (const v2f*)(X + (b * 128 + t + 1) * 256);
    if (full) fconv_kloop<true,  false>(kk, XM, XC, XP, 1.f, 1.f, W0, W1, W2, nb, acc0, acc1, acc2, acc3);
    else      fconv_kloop<false, false>(kk, XM, XC, XP, 1.f, 1.f, W0, W1, W2, nb, acc0, acc1, acc2, acc3);
  } else {
    float mm = (t > 0) ? 1.f : 0.f;
    float mp = (t < 127) ? 1.f : 0.f;
    int tm = t > 0 ? t - 1 : 0, tp = t < 127 ? t + 1 : 127;
    const v2f* XM = (const v2f*)(X + (b * 128 + tm) * 256);
    const v2f* XP = (const v2f*)(X + (b * 128 + tp) * 256);
    if (full) fconv_kloop<true,  true>(kk, XM, XC, XP, mm, mp, W0, W1, W2, nb, acc0, acc1, acc2, acc3);
    else      fconv_kloop<false, true>(kk, XM, XC, XP, mm, mp, W0, W1, W2, nb, acc0, acc1, acc2, acc3);
  }
  int mr0 = mTile * 16 + 8 * (l >> 4);
  int nc = strip * 64 + (l & 15);
#pragma unroll
  for (int v = 0; v < 8; v++) {
    float* dst = R + (mr0 + v) * NCOL;
    dst[nc] = acc0[v] + (nc < 514 ? bf[nc] : 0.f);
    if (full) {
      dst[nc + 16] = acc1[v] + (nc + 16 < 514 ? bf[nc + 16] : 0.f);
      dst[nc + 32] = acc2[v] + (nc + 32 < 514 ? bf[nc + 32] : 0.f);
      dst[nc + 48] = acc3[v] + (nc + 48 < 514 ? bf[nc + 48] : 0.f);
    }
  }
}

// ---------------- complex LTV recurrence, in place: R holds gain -> holds output -----------
__global__ void k_rec(float* __restrict__ R, const float* __restrict__ S) {
  int i = blockIdx.x * blockDim.x + threadIdx.x;
  if (i >= 1024 * 257) return;
  int b = i / 257, k = i % 257;
  float pr = 0.f, pi = 0.f;
  for (int f = 0; f < 128; f++) {
    int rr = (b * 128 + f) * NCOL;
    float gre = R[rr + k], gim = R[rr + 257 + k];
    float sre = 0.f, sim = 0.f;
    if (f < 64) { int sr = (b * 64 + f) * NCOL; sre = S[sr + k]; sim = S[sr + 257 + k]; }
    float nre = sre + pr * gre - pi * gim;
    float nim = sim + pr * gim + pi * gre;
    R[rr + k] = nre; R[rr + 257 + k] = nim;
    pr = nre; pi = nim;
  }
}

// ---------------- irfft GEMM [B*128,528]@[528,512] + fused overlap-add -----------------
__global__ void k_ifft_ola(const float* __restrict__ R, const float* __restrict__ D,
                           float* __restrict__ out) {
  const int NSTRIP = 8; // 512/64
  int wid = blockIdx.x * 4 + waveid();
  int mTile = wid / NSTRIP, strip = wid % NSTRIP;
  if (mTile >= 8192) return;
  bool safeStore = ((mTile & 7) != 7);      // only tiles containing f=127 can spill past 32768
  int l = threadIdx.x & 31, kk = (l >> 4) * 2;
  int mrow = mTile * 16 + (l & 15);
  int nb = strip * 64 + (l & 15);
  const v2f* A2 = (const v2f*)(R + (size_t)mrow * NCOL);
  const v2f* D2 = (const v2f*)D;
  v8f acc0 = {}, acc1 = {}, acc2 = {}, acc3 = {};
#pragma unroll 4
  for (int k0 = 0; k0 < NCOL; k0 += 4) {
    int kp = (k0 + kk) >> 1;
    v2f a = A2[kp];
    const v2f* dr = D2 + (size_t)kp * 512 + nb;
    acc0 = wmma4(a, dr[0],  acc0);
    acc1 = wmma4(a, dr[16], acc1);
    acc2 = wmma4(a, dr[32], acc2);
    acc3 = wmma4(a, dr[48], acc3);
  }
  int mr0 = mTile * 16 + 8 * (l >> 4);
  int nc = strip * 64 + (l & 15);
#pragma unroll
  for (int v = 0; v < 8; v++) {
    int r = mr0 + v;
    int b = r >> 7, f = r & 127;
    float* ob = out + b * 32768;
    int base = f * 256;                        // both frame halves land at n = 256f + j
    if (safeStore) {
      atomicAdd(ob + base + nc,      acc0[v]);
      atomicAdd(ob + base + nc + 16, acc1[v]);
      atomicAdd(ob + base + nc + 32, acc2[v]);
      atomicAdd(ob + base + nc + 48, acc3[v]);
    } else {
      int p0 = base + nc;      if (p0 < 32768) atomicAdd(ob + p0, acc0[v]);
      int p1 = base + nc + 16; if (p1 < 32768) atomicAdd(ob + p1, acc1[v]);
      int p2 = base + nc + 32; if (p2 < 32768) atomicAdd(ob + p2, acc2[v]);
      int p3 = base + nc + 48; if (p3 < 32768) atomicAdd(ob + p3, acc3[v]);
    }
  }
}

// ---------------- host launch ----------------
extern "C" void kernel_launch(void* const* d_in, const int* in_sizes, int n_in,
                              void* d_out, int out_size, void* d_ws, size_t ws_size,
                              hipStream_t stream) {
  (void)in_sizes; (void)n_in; (void)out_size; (void)ws_size;
  const float* latent  = (const float*)d_in[0];
  const float* impulse = (const float*)d_in[1];
  const float* w_lat   = (const float*)d_in[2];
  const float* b_lat   = (const float*)d_in[3];
  const float* ups_w   = (const float*)d_in[4];
  const float* bn_g    = (const float*)d_in[5];
  const float* bn_b    = (const float*)d_in[6];
  const float* fin_w   = (const float*)d_in[7];
  const float* fin_b   = (const float*)d_in[8];
  float* out = (float*)d_out;
  float* ws  = (float*)d_ws;

  float* Wstft = ws;                               // 512*528   (packed)
  float* Dmat  = Wstft + 512 * NCOL;               // 528*512   (packed)
  float* PWlat = Dmat + NCOL * 512;                // 256*2048  (packed)
  float* PUps  = PWlat + 256 * 2048;               // 16 x 65536 (packed, [layer*4+tap])
  float* PFin  = PUps + 16 * 65536;                // 3 x 135168 (packed, zero-padded N=528)
  float* Spec  = PFin + 3 * 128 * NCOL * 2;        // 1024*64*528
  float* B0    = Spec + 1024 * 64 * NCOL;          // 1024*128*256
  float* B1    = B0 + 1024 * 128 * 256;            // 1024*128*256
  float* Res   = B1 + 1024 * 128 * 256;            // 1024*128*528
  float* Stats = Res + 1024 * 128 * NCOL;          // 512

  k_zero<<<(33554432 + 255) / 256, 256, 0, stream>>>(out, 33554432);
  k_init_mats<<<(2 * 512 * NCOL + 255) / 256, 256, 0, stream>>>(Wstft, Dmat);
  k_pack_wlat<<<(256 * 2048 + 255) / 256, 256, 0, stream>>>(w_lat, PWlat);
  k_pack_ups<<<(4 * 256 * 256 * 4 + 255) / 256, 256, 0, stream>>>(ups_w, PUps);
  k_pack_fin<<<(3 * 128 * NCOL * 2 + 255) / 256, 256, 0, stream>>>(fin_w, PFin);

  { int waves = 4096 * 9; k_stft<<<(waves + 3) / 4, 128, 0, stream>>>(impulse, Wstft, Spec); }
  { int waves = 64 * 32;  k_lat<<<(waves + 3) / 4, 128, 0, stream>>>(latent, PWlat, b_lat, B0); }

  float* xin = B0; float* xout = B1;
  for (int layer = 0; layer < 4; layer++) {
    int lg = 3 + layer;
    int Lin = 1 << lg, Lout = 2 * Lin;
    const float* T0 = PUps + (layer * 4 + 0) * 65536;
    const float* T1 = PUps + (layer * 4 + 1) * 65536;
    const float* T2 = PUps + (layer * 4 + 2) * 65536;
    const float* T3 = PUps + (layer * 4 + 3) * 65536;
    int waves = ((1024 * Lin) >> 4) * 4;
    k_convt<<<(waves + 3) / 4, 128, 0, stream>>>(xin, T0, T2, xout, lg, 0); // even: taps 0,2
    k_convt<<<(waves + 3) / 4, 128, 0, stream>>>(xin, T1, T3, xout, lg, 1); // odd:  taps 1,3
    int rows = 1024 * Lout;
    k_zero<<<2, 256, 0, stream>>>(Stats, 512);
    k_bnreduce<<<(rows + 127) / 128, 256, 0, stream>>>(xout, Stats, rows);
    k_bnapply<<<(rows * 256 + 255) / 256, 256, 0, stream>>>(
        xout, Stats, bn_g + layer * 256, bn_b + layer * 256, rows * 256, 1.f / (float)rows);
    float* tmp = xin; xin = xout; xout = tmp;
  }
  // xin now holds [B,128,256] activations
  { int waves = 8192 * 9; k_fconv<<<(waves + 3) / 4, 128, 0, stream>>>(xin, PFin, fin_b, Res); }
  k_rec<<<(1024 * 257 + 255) / 256, 256, 0, stream>>>(Res, Spec);
  { int waves = 8192 * 8; k_ifft_ola<<<(waves + 3) / 4, 128, 0, stream>>>(Res, Dmat, out); }
}